// phase_transforms_40243843564154
// MI455X (gfx1250) — compile-verified
//
#include <hip/hip_runtime.h>
#include <math.h>

// Problem constants (from the reference).
#define NDIM  4096
#define BATCH 16384

// Tiling: 256 threads x 4 cols = 1024-wide column tile; 16 rows per block.
#define TPB            256
#define COLS_PER_T     4
#define TILE_W         (TPB * COLS_PER_T)   // 1024
#define ROWS_PER_BLOCK 16
#define DEPTH          4                    // async pipeline depth (32 KB LDS)

typedef float f32x4 __attribute__((ext_vector_type(4)));
typedef int   v4i   __attribute__((ext_vector_type(4)));   // builtin param type

static __device__ __forceinline__ void wait_asynccnt_6() {
#if __has_builtin(__builtin_amdgcn_s_wait_asynccnt)
    __builtin_amdgcn_s_wait_asynccnt(2 * (DEPTH - 1));
#else
    asm volatile("s_wait_asynccnt 6" ::: "memory");
#endif
}

static __device__ __forceinline__ void wait_dscnt_0() {
    asm volatile("s_wait_dscnt 0" ::: "memory");
}

// Issue one async 128-bit global->LDS copy (per active lane).
static __device__ __forceinline__ void async_copy_b128(const float* g, float* l) {
    __builtin_amdgcn_global_load_async_to_lds_b128((v4i*)g, (v4i*)l, 0, 0);
}

__global__ __launch_bounds__(TPB)
void phase_rot_40243843564154_kernel(const float* __restrict__ x_re,
                                     const float* __restrict__ x_im,
                                     const float* __restrict__ phases,
                                     float* __restrict__ out)
{
    // Stage layout: [stage][re: TILE_W floats][im: TILE_W floats]
    __shared__ __align__(16) float lds[DEPTH * 2 * TILE_W];   // 32 KB

    const int       t    = threadIdx.x;
    const int       col  = blockIdx.x * TILE_W + t * COLS_PER_T;
    const long long row0 = (long long)blockIdx.y * ROWS_PER_BLOCK;
    const long long base = row0 * NDIM + col;

    const float* gre = x_re + base;
    const float* gim = x_im + base;
    float*       ore = out + base;                              // out[0] = real
    float*       oim = out + (long long)BATCH * NDIM + base;    // out[1] = imag

    // Per-thread rotation coefficients: one float4 load of phases + 4 sincos,
    // amortized over ROWS_PER_BLOCK rows (trig cost negligible vs 1 GiB HBM).
    const f32x4 ph = *(const f32x4*)(phases + col);
    f32x4 c, s;
#pragma unroll
    for (int k = 0; k < COLS_PER_T; ++k) {
        float ss, cc;
        sincosf(ph[k], &ss, &cc);
        s[k] = ss; c[k] = cc;
    }

    // ---- gfx1250 async global->LDS pipeline (ASYNCcnt), depth 4 ----
    // Each lane stages its own 16B of re and 16B of im per row into
    // lane-private LDS slots; no barriers needed.
#pragma unroll
    for (int p = 0; p < DEPTH; ++p) {
        float* slot = lds + p * 2 * TILE_W;
        async_copy_b128(gre + (long long)p * NDIM, slot + t * 4);
        async_copy_b128(gim + (long long)p * NDIM, slot + TILE_W + t * 4);
    }

#pragma unroll
    for (int i = 0; i < ROWS_PER_BLOCK; ++i) {
        // Async loads complete in order: outstanding <= 2*(DEPTH-1) implies
        // stage i's two loads have landed in LDS only for i <= ROWS_PER_BLOCK-DEPTH; the last DEPTH-1 rows are not waited for.
        wait_asynccnt_6();

        float* slot = lds + (i % DEPTH) * 2 * TILE_W;
        f32x4 re = *(const f32x4*)(slot + t * 4);           // ds_load_b128
        f32x4 im = *(const f32x4*)(slot + TILE_W + t * 4);  // ds_load_b128

        f32x4 o_re = re * c - im * s;
        f32x4 o_im = re * s + im * c;

        // DS reads vs async LDS writes use different counters and are
        // unordered: drain DScnt before recycling this slot.
        wait_dscnt_0();

        const int nr = i + DEPTH;
        if (nr < ROWS_PER_BLOCK) {
            async_copy_b128(gre + (long long)nr * NDIM, slot + t * 4);
            async_copy_b128(gim + (long long)nr * NDIM, slot + TILE_W + t * 4);
        }

        // Streaming output: written once, never re-read -> non-temporal B128.
        __builtin_nontemporal_store(o_re, (f32x4*)(ore + (long long)i * NDIM));
        __builtin_nontemporal_store(o_im, (f32x4*)(oim + (long long)i * NDIM));
    }
}

extern "C" void kernel_launch(void* const* d_in, const int* in_sizes, int n_in,
                              void* d_out, int out_size, void* d_ws, size_t ws_size,
                              hipStream_t stream) {
    const float* x_re   = (const float*)d_in[0];
    const float* x_im   = (const float*)d_in[1];
    const float* phases = (const float*)d_in[2];
    float*       out    = (float*)d_out;

    dim3 grid(NDIM / TILE_W, BATCH / ROWS_PER_BLOCK);   // (4, 1024)
    phase_rot_40243843564154_kernel<<<grid, TPB, 0, stream>>>(x_re, x_im, phases, out);
}